// Spectral_18940805775443
// MI455X (gfx1250) — compile-verified
//
#include <hip/hip_runtime.h>
#include <hip/hip_bf16.h>

// ---------------------------------------------------------------------------
// Spectral conv: out = relu( eig^T @ (param ⊙_e (square^T @ eig^T)) @ eig )
// BS=16 T=16 N=512 C=128 E=128 OUT=256, P=T*N=8192, bt = b*16+t (256 pairs)
// Stage 1: x1[bt,c,e]   = sum_n feats[b, n*16+t, c] * eig[e,n]      (GEMM K=512)
// Stage 2: x3[bt,e,o]   = sum_c x1[bt,c,e] * param[o,c,e]           (per-e GEMM)
// Stage 3: out[bt,n,o]  = relu( sum_e x3[bt,e,o] * eig[e,n] )       (GEMM K=128)
// All matrix math on v_wmma_f32_16x16x4_f32 (fp32, matches reference dtype).
// ---------------------------------------------------------------------------

typedef float v2f __attribute__((ext_vector_type(2)));
typedef float v8f __attribute__((ext_vector_type(8)));

#define LDS_K 16
#define LDS_STRIDE 18   // 16 K-values + 2 pad floats -> conflict-free, 8B aligned

// As: [M=128][K=16] (stride 18), Bs: [Ncol=128][K=16] (stride 18).
// Wave computes rows [mbase, mbase+16) x 128 cols as 8 accumulator tiles.
// WMMA 16x16x4 fragment layout (ISA 7.12.2):
//   A: lane L<16 holds (M=L, K=kk..kk+1), lane L+16 holds (M=L, K=kk+2..kk+3)
//   B: lane L<16 holds (K=kk..kk+1, N=L), lane L+16 holds (K=kk+2..kk+3, N=L)
__device__ __forceinline__ void mma_chunk(const float* As, const float* Bs,
                                          v8f acc[8], int mbase, int lane) {
  const int nL   = lane & 15;
  const int koff = (lane < 16) ? 0 : 2;
#pragma unroll
  for (int kk = 0; kk < LDS_K; kk += 4) {
    v2f a = *(const v2f*)(As + (mbase + nL) * LDS_STRIDE + kk + koff);
#pragma unroll
    for (int et = 0; et < 8; ++et) {
      v2f b = *(const v2f*)(Bs + (et * 16 + nL) * LDS_STRIDE + kk + koff);
      acc[et] = __builtin_amdgcn_wmma_f32_16x16x4_f32(
          false, a, false, b, (short)0, acc[et], false, false);
    }
  }
}

// ---------------- k0: param (O=256,C=128,E=128) -> paramT (E,C,O) -----------
__global__ __launch_bounds__(256) void k0_transpose(const float* __restrict__ param,
                                                    float* __restrict__ paramT) {
  __shared__ float tile[32][33];
  const int c     = blockIdx.x >> 5;        // 0..127
  const int obt   = (blockIdx.x >> 2) & 7;  // 0..7  (O/32)
  const int ebt   = blockIdx.x & 3;         // 0..3  (E/32)
  const int tid   = threadIdx.x;
  const int obase = obt * 32, ebase = ebt * 32;
#pragma unroll
  for (int i = 0; i < 4; ++i) {
    int idx = i * 256 + tid;
    int jo = idx >> 5, je = idx & 31;       // je inner -> coalesced in E
    tile[jo][je] = param[((size_t)(obase + jo) * 128 + c) * 128 + ebase + je];
  }
  __syncthreads();
#pragma unroll
  for (int i = 0; i < 4; ++i) {
    int idx = i * 256 + tid;
    int je = idx >> 5, jo = idx & 31;       // jo inner -> coalesced in O
    paramT[((size_t)(ebase + je) * 128 + c) * 256 + obase + jo] = tile[jo][je];
  }
}

// ---------------- k1: x1[bt,c,e] = sum_n feats[b,n*16+t,c] * eig[e,n] -------
// grid.x = 256 (one workgroup per bt). M=c(128), Ncol=e(128), K=n(512).
__global__ __launch_bounds__(256) void k1_x1(const float* __restrict__ feats,
                                             const float* __restrict__ eig,
                                             float* __restrict__ x1w) {
  __shared__ float As[128 * LDS_STRIDE];
  __shared__ float Bs[128 * LDS_STRIDE];
  const int bt = blockIdx.x;
  const int b = bt >> 4, t = bt & 15;
  const int tid = threadIdx.x, lane = tid & 31, w = tid >> 5;
  const float* fb = feats + (size_t)b * (8192 * 128);

  v8f acc[8];
  const v8f vzero = {0.f, 0.f, 0.f, 0.f, 0.f, 0.f, 0.f, 0.f};
#pragma unroll
  for (int i = 0; i < 8; ++i) acc[i] = vzero;

  for (int nb = 0; nb < 32; ++nb) {
    const int nbase = nb * 16;
    __syncthreads();
    // A chunk: As[c][nn] = feats[b, (nbase+nn)*16+t, c]   (c inner -> coalesced)
#pragma unroll
    for (int i = 0; i < 8; ++i) {
      int idx = i * 256 + tid;
      int nn = idx >> 7, c = idx & 127;
      As[c * LDS_STRIDE + nn] = fb[((size_t)(nbase + nn) * 16 + t) * 128 + c];
    }
    // B chunk: Bs[e][nn] = eig[e, nbase+nn]
#pragma unroll
    for (int i = 0; i < 8; ++i) {
      int idx = i * 256 + tid;
      int e = idx >> 4, nn = idx & 15;
      Bs[e * LDS_STRIDE + nn] = eig[(size_t)e * 512 + nbase + nn];
    }
    if (nb + 1 < 32)  // hint next feats chunk toward L2 (global_prefetch_b8)
      __builtin_prefetch(fb + ((size_t)(nbase + 16) * 16 + t) * 128 + (tid & 127), 0, 1);
    __syncthreads();
    mma_chunk(As, Bs, acc, w * 16, lane);
  }

  // Store x1w[bt*16384 + c*128 + e]; lanes 0-15 / 16-31 are contiguous in e.
  float* op = x1w + (size_t)bt * 16384;
  const int nL = lane & 15, hi = (lane >= 16) ? 8 : 0;
#pragma unroll
  for (int et = 0; et < 8; ++et)
#pragma unroll
    for (int r = 0; r < 8; ++r)
      op[(w * 16 + r + hi) * 128 + et * 16 + nL] = acc[et][r];
}

// ---------------- k2: x3[bt,e,o] = sum_c x1[bt,c,e] * paramT[e,c,o] ---------
// grid.x = 128*2*2 = 512. Per block: fixed e, M=bt tile 128, Ncol=o tile 128, K=c=128.
__global__ __launch_bounds__(256) void k2_x3(const float* __restrict__ x1w,
                                             const float* __restrict__ paramT,
                                             float* __restrict__ x3w) {
  __shared__ float As[128 * LDS_STRIDE];
  __shared__ float Bs[128 * LDS_STRIDE];
  const int e  = blockIdx.x >> 2;
  const int mt = (blockIdx.x >> 1) & 1;
  const int nt = blockIdx.x & 1;
  const int tid = threadIdx.x, lane = tid & 31, w = tid >> 5;
  const int mbase = mt * 128, nbase = nt * 128;

  v8f acc[8];
  const v8f vzero = {0.f, 0.f, 0.f, 0.f, 0.f, 0.f, 0.f, 0.f};
#pragma unroll
  for (int i = 0; i < 8; ++i) acc[i] = vzero;

  for (int cb = 0; cb < 128; cb += 16) {
    __syncthreads();
    // A chunk: As[m][k] = x1w[(mbase+m)*16384 + (cb+k)*128 + e]  (L2-resident gather)
#pragma unroll
    for (int i = 0; i < 8; ++i) {
      int idx = i * 256 + tid;
      int m = idx >> 4, k = idx & 15;
      As[m * LDS_STRIDE + k] =
          x1w[(size_t)(mbase + m) * 16384 + (cb + k) * 128 + e];
    }
    // B chunk: Bs[o][k] = paramT[(e*128 + cb+k)*256 + nbase+o]  (o inner -> coalesced)
#pragma unroll
    for (int i = 0; i < 8; ++i) {
      int idx = i * 256 + tid;
      int k = idx >> 7, o = idx & 127;
      Bs[o * LDS_STRIDE + k] =
          paramT[((size_t)e * 128 + cb + k) * 256 + nbase + o];
    }
    __syncthreads();
    mma_chunk(As, Bs, acc, w * 16, lane);
  }

  // Store x3w[bt*32768 + e*256 + o]; contiguous in o per half-row.
  const int nL = lane & 15, hi = (lane >= 16) ? 8 : 0;
#pragma unroll
  for (int et = 0; et < 8; ++et)
#pragma unroll
    for (int r = 0; r < 8; ++r) {
      int btv = mbase + w * 16 + r + hi;
      int o   = nbase + et * 16 + nL;
      x3w[(size_t)btv * 32768 + e * 256 + o] = acc[et][r];
    }
}

// ---------------- k3: out[bt,n,o] = relu(sum_e eig[e,n] * x3[bt,e,o]) -------
// grid.x = 256*4*2 = 2048. Per block: fixed bt, M=n tile 128, Ncol=o tile 128, K=e=128.
__global__ __launch_bounds__(256) void k3_out(const float* __restrict__ eig,
                                              const float* __restrict__ x3w,
                                              float* __restrict__ out) {
  __shared__ float As[128 * LDS_STRIDE];
  __shared__ float Bs[128 * LDS_STRIDE];
  const int bt  = blockIdx.x >> 3;
  const int nt4 = (blockIdx.x >> 1) & 3;
  const int ot  = blockIdx.x & 1;
  const int tid = threadIdx.x, lane = tid & 31, w = tid >> 5;

  v8f acc[8];
  const v8f vzero = {0.f, 0.f, 0.f, 0.f, 0.f, 0.f, 0.f, 0.f};
#pragma unroll
  for (int i = 0; i < 8; ++i) acc[i] = vzero;

  for (int eb = 0; eb < 128; eb += 16) {
    __syncthreads();
    // A chunk: As[nloc][k] = eig[(eb+k)*512 + nt4*128 + nloc]  (nloc inner -> coalesced)
#pragma unroll
    for (int i = 0; i < 8; ++i) {
      int idx = i * 256 + tid;
      int k = idx >> 7, nloc = idx & 127;
      As[nloc * LDS_STRIDE + k] = eig[(size_t)(eb + k) * 512 + nt4 * 128 + nloc];
    }
    // B chunk: Bs[o][k] = x3w[bt*32768 + (eb+k)*256 + ot*128 + o]
#pragma unroll
    for (int i = 0; i < 8; ++i) {
      int idx = i * 256 + tid;
      int k = idx >> 7, o = idx & 127;
      Bs[o * LDS_STRIDE + k] =
          x3w[(size_t)bt * 32768 + (eb + k) * 256 + ot * 128 + o];
    }
    __syncthreads();
    mma_chunk(As, Bs, acc, w * 16, lane);
  }

  // out[(bt*512 + n)*256 + o] with fused ReLU; contiguous in o per half-row.
  const int nL = lane & 15, hi = (lane >= 16) ? 8 : 0;
#pragma unroll
  for (int et = 0; et < 8; ++et)
#pragma unroll
    for (int r = 0; r < 8; ++r) {
      int nfull = nt4 * 128 + w * 16 + r + hi;
      int ofull = ot * 128 + et * 16 + nL;
      out[((size_t)bt * 512 + nfull) * 256 + ofull] = fmaxf(acc[et][r], 0.f);
    }
}

// ---------------------------------------------------------------------------
extern "C" void kernel_launch(void* const* d_in, const int* in_sizes, int n_in,
                              void* d_out, int out_size, void* d_ws, size_t ws_size,
                              hipStream_t stream) {
  // setup_inputs order: feats, pos, times, ids, eig, param
  const float* feats = (const float*)d_in[0];
  const float* eig   = (const float*)d_in[4];
  const float* param = (const float*)d_in[5];
  float* out = (float*)d_out;

  // Workspace layout (floats): x1 (256*128*128) | x3 (256*128*256) | paramT (128*128*256)
  float* x1w    = (float*)d_ws;
  float* x3w    = x1w + (size_t)256 * 128 * 128;   // + 4,194,304
  float* paramT = x3w + (size_t)256 * 128 * 256;   // + 8,388,608  (total 64 MiB)

  k0_transpose<<<4096, 256, 0, stream>>>(param, paramT);
  k1_x1<<<256, 256, 0, stream>>>(feats, eig, x1w);
  k2_x3<<<512, 256, 0, stream>>>(x1w, paramT, x3w);
  k3_out<<<2048, 256, 0, stream>>>(eig, x3w, out);
}